// ChebConv_14499809591384
// MI455X (gfx1250) — compile-verified
//
#include <hip/hip_runtime.h>
#include <hip/hip_bf16.h>

typedef __attribute__((ext_vector_type(2))) float v2f;
typedef __attribute__((ext_vector_type(8))) float v8f;

#define CH 64            // channels (IN_C == OUT_C == 64)
#define NB 2             // batch

// ---------------------------------------------------------------- utilities
__global__ void zero_f32_kernel(float* __restrict__ p, long n) {
    long i = (long)blockIdx.x * blockDim.x + threadIdx.x;
    long stride = (long)gridDim.x * blockDim.x;
    for (; i < n; i += stride) p[i] = 0.0f;
}

// degree of each source node, excluding self loops
__global__ void degree_kernel(const int* __restrict__ row, const int* __restrict__ col,
                              float* __restrict__ deg, long E) {
    long e = (long)blockIdx.x * blockDim.x + threadIdx.x;
    if (e >= E) return;
    int r = row[e], c = col[e];
    if (r != c) atomicAdd(&deg[r], 1.0f);
}

// deg -> deg^{-1/2} in place
__global__ void dinv_sqrt_kernel(float* __restrict__ deg, long n) {
    long i = (long)blockIdx.x * blockDim.x + threadIdx.x;
    if (i >= n) return;
    float d = deg[i];
    deg[i] = (d > 0.0f) ? __frsqrt_rn(d) : 0.0f;
}

// lap[e] = -dis[row]*ew*dis[col]  (0 for self loops)
__global__ void lap_kernel(const int* __restrict__ row, const int* __restrict__ col,
                           const float* __restrict__ ew, const float* __restrict__ dis,
                           float* __restrict__ lap, long E) {
    long e = (long)blockIdx.x * blockDim.x + threadIdx.x;
    if (e >= E) return;
    int r = row[e], c = col[e];
    lap[e] = (r != c) ? (-dis[r] * ew[e] * dis[c]) : 0.0f;
}

// dst[b, row[e], c] += lap[e] * src[b, col[e], c]
// one thread per (edge, batch, channel); 128 work items per edge, coalesced in c.
// This is the HBM-roofline-dominant phase (~1.6 GB gather+scatter total).
__global__ __launch_bounds__(256) void spmm_kernel(
    const float* __restrict__ src, const int* __restrict__ row,
    const int* __restrict__ col, const float* __restrict__ lap,
    float* __restrict__ dst, int N, long E) {
    long idx = (long)blockIdx.x * blockDim.x + threadIdx.x;
    long e = idx >> 7;               // / (NB*CH) = /128
    if (e >= E) return;
    int rem = (int)(idx & 127);
    int b = rem >> 6;
    int c = rem & 63;
    float l = lap[e];
    if (l == 0.0f) return;           // self loops contribute nothing
    long cs = col[e], rs = row[e];
    float v = src[((long)b * N + cs) * CH + c] * l;
    atomicAdd(&dst[((long)b * N + rs) * CH + c], v);
}

// ---------------------------------------------------------------- WMMA GEMM
// out[m, :] = x[m,:]@W0 + tx1[m,:]@W1 + (2*s2[m,:] - x[m,:])@W2 + bias
// One wave per 16-row tile of M = B*N rows; 4 waves (128 threads) per block.
// fp32 WMMA 16x16x4: exact precision; GEMM is far from the bottleneck.
//
// LDS weight layout is pair-interleaved so each B fragment is ONE aligned
// ds_load_b64 landing directly in a contiguous even-aligned VGPR pair:
//   pair index p(t, j, n) = (t*32 + j)*64 + n   holds (W_t[2j][n], W_t[2j+1][n])
__global__ __launch_bounds__(128) void cheb_gemm_wmma(
    const float* __restrict__ x, const float* __restrict__ tx1,
    const float* __restrict__ s2, const float* __restrict__ w,
    const float* __restrict__ bias, float* __restrict__ out, long BN) {
    __shared__ float wlds[3 * CH * CH];   // 48 KB, viewed as 6144 float2 pairs

    const int tid = threadIdx.x;
    // cooperative load + pair-interleave: global reads stay coalesced
    for (int idx = tid; idx < 3 * CH * CH; idx += 128) {
        const int t   = idx >> 12;          // /4096
        const int rem = idx & 4095;
        const int k   = rem >> 6;           // row (K)
        const int n   = rem & 63;           // col (N)
        wlds[((((t << 5) + (k >> 1)) << 6) + n) * 2 + (k & 1)] = w[idx];
    }
    __syncthreads();

    const int wave = tid >> 5;
    const int lane = tid & 31;
    const long m0 = ((long)blockIdx.x * 4 + wave) * 16;
    if (m0 + 16 > BN) return;            // whole-wave guard: EXEC stays all-ones

    const int mrow = lane & 15;           // A: row M; B: col N (within tile)
    const int lhi  = lane >> 4;           // 0 or 1 (selects K half-pair)

    // per-lane base into the pair array; all (t,ks,nt) offsets are immediates
    const v2f* __restrict__ wp = (const v2f*)wlds;
    const v2f* __restrict__ bbase = wp + (lhi << 6) + mrow;   // + t*2048 + ks*128 + nt*16, see below

    v8f acc[4] = {v8f{}, v8f{}, v8f{}, v8f{}};

    #pragma unroll
    for (int t = 0; t < 3; ++t) {
        const float* A = (t == 0) ? x : (t == 1) ? tx1 : s2;
        #pragma unroll
        for (int ks = 0; ks < 16; ++ks) {
            const int kb = ks * 4 + (lhi << 1);
            // A fragment: lane holds A[m0+mrow][kb], A[m0+mrow][kb+1]  (one b64)
            const float2 a2 = *(const float2*)(A + (m0 + mrow) * CH + kb);
            v2f a;
            a.x = a2.x; a.y = a2.y;
            if (t == 2) {                 // fuse Tx2 = 2*spmm(Tx1) - Tx0
                const float2 x2 = *(const float2*)(x + (m0 + mrow) * CH + kb);
                a.x = 2.0f * a.x - x2.x;
                a.y = 2.0f * a.y - x2.y;
            }
            #pragma unroll
            for (int nt = 0; nt < 4; ++nt) {
                // pair j = ks*2 + lhi, n = nt*16 + mrow  ->  one ds_load_b64
                const v2f b = bbase[(t * 32 + ks * 2) * 64 + nt * 16];
                acc[nt] = __builtin_amdgcn_wmma_f32_16x16x4_f32(
                    false, a, false, b, (short)0, acc[nt], false, false);
            }
        }
    }

    // D layout: VGPR v holds rows m0+v (lanes 0-15) and m0+v+8 (lanes 16-31)
    const int moff = lhi << 3;            // 0 or 8
    #pragma unroll
    for (int nt = 0; nt < 4; ++nt) {
        const int col = nt * 16 + mrow;
        const float bv = bias[col];
        #pragma unroll
        for (int v = 0; v < 8; ++v) {
            out[(m0 + moff + v) * CH + col] = acc[nt][v] + bv;
        }
    }
}

// scalar tail for BN % 16 rows (not taken for the reference shape)
__global__ void cheb_gemm_tail(const float* __restrict__ x, const float* __restrict__ tx1,
                               const float* __restrict__ s2, const float* __restrict__ w,
                               const float* __restrict__ bias, float* __restrict__ out,
                               long startRow, long BN) {
    long r = startRow + blockIdx.x;
    if (r >= BN) return;
    int c = threadIdx.x;   // 64 threads
    float acc = bias[c];
    for (int k = 0; k < CH; ++k) {
        float t0 = x[r * CH + k];
        float t1 = tx1[r * CH + k];
        float t2 = 2.0f * s2[r * CH + k] - t0;
        acc += t0 * w[k * CH + c] + t1 * w[CH * CH + k * CH + c] + t2 * w[2 * CH * CH + k * CH + c];
    }
    out[r * CH + c] = acc;
}

// ---------------------------------------------------------------- launcher
extern "C" void kernel_launch(void* const* d_in, const int* in_sizes, int n_in,
                              void* d_out, int out_size, void* d_ws, size_t ws_size,
                              hipStream_t stream) {
    const float* x    = (const float*)d_in[0];
    const int*   ei   = (const int*)d_in[1];
    const float* ew   = (const float*)d_in[2];
    const float* w    = (const float*)d_in[3];
    const float* bias = (const float*)d_in[4];
    float*       out  = (float*)d_out;

    const long BN = (long)in_sizes[0] / CH;   // B*N rows
    const int  N  = (int)(BN / NB);
    const long E  = (long)in_sizes[2];
    const int* row = ei;
    const int* col = ei + E;

    // workspace layout (256B aligned regions)
    char* ws = (char*)d_ws;
    size_t off = 0;
    auto carve = [&](size_t bytes) {
        void* p = ws + off;
        off += (bytes + 255) & ~(size_t)255;
        return p;
    };
    float* deg = (float*)carve((size_t)N * sizeof(float));
    float* lap = (float*)carve((size_t)E * sizeof(float));
    float* tx1 = (float*)carve((size_t)BN * CH * sizeof(float));
    float* s2  = (float*)carve((size_t)BN * CH * sizeof(float));
    (void)ws_size;

    const int TB = 256;
    auto gblocks = [](long n, int tb) { return (int)((n + tb - 1) / tb); };

    // zero accumulation buffers (grid-stride; bounded grids)
    zero_f32_kernel<<<4096, TB, 0, stream>>>(deg, N);
    zero_f32_kernel<<<4096, TB, 0, stream>>>(tx1, BN * CH);
    zero_f32_kernel<<<4096, TB, 0, stream>>>(s2,  BN * CH);

    // normalized Laplacian coefficients
    degree_kernel<<<gblocks(E, TB), TB, 0, stream>>>(row, col, deg, E);
    dinv_sqrt_kernel<<<gblocks(N, TB), TB, 0, stream>>>(deg, N);
    lap_kernel<<<gblocks(E, TB), TB, 0, stream>>>(row, col, ew, deg, lap, E);

    // Tx1 = spmm(x);  s2 = spmm(Tx1)
    const long spmmThreads = E * (long)(NB * CH);
    spmm_kernel<<<gblocks(spmmThreads, TB), TB, 0, stream>>>(x,   row, col, lap, tx1, N, E);
    spmm_kernel<<<gblocks(spmmThreads, TB), TB, 0, stream>>>(tx1, row, col, lap, s2,  N, E);

    // fused 3-term GEMM + bias via fp32 WMMA
    const long fullTiles = BN / 16;
    if (fullTiles > 0) {
        const int blocks = (int)((fullTiles + 3) / 4);
        cheb_gemm_wmma<<<blocks, 128, 0, stream>>>(x, tx1, s2, w, bias, out, BN);
    }
    const long tailStart = fullTiles * 16;
    if (tailStart < BN) {
        cheb_gemm_tail<<<(int)(BN - tailStart), CH, 0, stream>>>(x, tx1, s2, w, bias, out,
                                                                 tailStart, BN);
    }
}